// SAMPrimeSimulator_16681652978224
// MI455X (gfx1250) — compile-verified
//
#include <hip/hip_runtime.h>
#include <hip/hip_bf16.h>
#include <math.h>

// ---------------------------------------------------------------------------
// SAMPrimeSimulator for gfx1250 (MI455X, wave32).
// Pass 1: reduce sum(g^2), sum(hvp^2)   (streams 4 arrays, 256 MB)
// Pass 2: finalize scalars (1 block)
// Pass 3: elementwise output            (streams 2 arrays + out, 192 MB)
// Block-level reductions use V_WMMA_F32_16X16X4_F32 (A = ones -> column sums).
// ---------------------------------------------------------------------------

typedef __attribute__((ext_vector_type(2))) float v2f;
typedef __attribute__((ext_vector_type(8))) float v8f;

#define EPSF      1e-12f
#define RHO_THR   (0.62f * 1.08f)      // RHO_FLAT * (1 + HYST)
#define T_ENT     0.074f
#define INV_SQRTN (1.0f / 4096.0f)     // 1/sqrt(16*1024*1024)
#define SQ1MC     0.9999995f           // sqrt(1 - 1e-6)
#define SQC       0.001f               // sqrt(1e-6)
#define TWO_PI    6.2831853071795864f

#define NBLK_RED  4096                 // blocks in reduce pass (partial pairs)
#define TOTALS_OFF (2 * NBLK_RED)      // ws float offset of the two totals

__device__ __forceinline__ float wave_sum(float v) {
  // wave32 butterfly; all lanes end with the full sum
  #pragma unroll
  for (int m = 16; m > 0; m >>= 1) v += __shfl_xor(v, m, 32);
  return v;
}

// Reduce (sg, sh) across a 256-thread block (8 waves) and write {sum_g, sum_h}
// to dst[0..1]. Final 8+8 partials are collapsed with one
// V_WMMA_F32_16X16X4_F32: A = ones(16x4), B(4x16) carries partials in
// columns 0..3, so D[0][n] = column-sum. Wave 0 executes the WMMA with EXEC
// all-ones (uniform branch over the whole wave).
__device__ __forceinline__ void block_reduce_pair(float sg, float sh, float* dst) {
  __shared__ float lds[16];
  const int lane = threadIdx.x & 31;
  const int wv   = threadIdx.x >> 5;

  sg = wave_sum(sg);
  sh = wave_sum(sh);
  if (lane == 0) { lds[wv] = sg; lds[8 + wv] = sh; }  // pg[w]=lds[w], ph[w]=lds[8+w]
  __syncthreads();

  if (threadIdx.x < 32) {                 // wave 0 only, EXEC = all ones
    const int n    = lane & 15;           // B column
    const int half = lane >> 4;           // which B-row pair this lane carries
    // B VGPR0 holds rows k={0,1}, VGPR1 rows k={2,3}; lanes 0-15 = first row
    // of the pair, lanes 16-31 = second. Column n packs lds[4n+k].
    float b0 = 0.0f, b1 = 0.0f;
    if (n < 4) {
      b0 = lds[4 * n + half];             // k = half
      b1 = lds[4 * n + 2 + half];         // k = 2 + half
    }
    v2f a; a.x = 1.0f; a.y = 1.0f;        // A = ones(16x4)
    v2f b; b.x = b0;   b.y = b1;
    v8f acc = {};
    acc = __builtin_amdgcn_wmma_f32_16x16x4_f32(
        /*neg_a=*/false, a, /*neg_b=*/false, b,
        /*c_mod=*/(short)0, acc, /*reuse_a=*/false, /*reuse_b=*/false);
    // D row 0 sits in acc[0], lanes 0..15 = columns 0..15.
    const float d0 = acc[0];
    const float tg = __shfl(d0, 0, 32) + __shfl(d0, 1, 32);  // pg[0..3]+pg[4..7]
    const float th = __shfl(d0, 2, 32) + __shfl(d0, 3, 32);  // ph[0..3]+ph[4..7]
    if (lane == 0) { dst[0] = tg; dst[1] = th; }
  }
}

__device__ __forceinline__ float hvp_term(float dg, float dt) {
  return (fabsf(dt) > EPSF) ? (dg / (dt + EPSF)) : 0.0f;
}

// ---------------- Pass 1: per-block partial sums of squares ----------------
__global__ __launch_bounds__(256)
void k_reduce(const float* __restrict__ g, const float* __restrict__ gp,
              const float* __restrict__ tc, const float* __restrict__ tp,
              float* __restrict__ partials, int n4, int n) {
  const int tid    = blockIdx.x * blockDim.x + threadIdx.x;
  const int stride = gridDim.x * blockDim.x;
  const float4* g4  = (const float4*)g;
  const float4* gp4 = (const float4*)gp;
  const float4* tc4 = (const float4*)tc;
  const float4* tp4 = (const float4*)tp;

  float sg = 0.0f, sh = 0.0f;
  for (int i = tid; i < n4; i += stride) {
    if (i + stride < n4) {                      // global_prefetch_b8 next tile
      __builtin_prefetch(&g4[i + stride],  0, 0);
      __builtin_prefetch(&gp4[i + stride], 0, 0);
      __builtin_prefetch(&tc4[i + stride], 0, 0);
      __builtin_prefetch(&tp4[i + stride], 0, 0);
    }
    const float4 a = g4[i];
    const float4 b = gp4[i];
    const float4 c = tc4[i];
    const float4 d = tp4[i];
    sg += a.x * a.x + a.y * a.y + a.z * a.z + a.w * a.w;
    const float hx = hvp_term(a.x - b.x, c.x - d.x);
    const float hy = hvp_term(a.y - b.y, c.y - d.y);
    const float hz = hvp_term(a.z - b.z, c.z - d.z);
    const float hw = hvp_term(a.w - b.w, c.w - d.w);
    sh += hx * hx + hy * hy + hz * hz + hw * hw;
  }
  // scalar tail (N not divisible by 4)
  for (int i = 4 * n4 + tid; i < n; i += stride) {
    const float a = g[i];
    const float h = hvp_term(a - gp[i], tc[i] - tp[i]);
    sg += a * a;
    sh += h * h;
  }
  block_reduce_pair(sg, sh, &partials[2 * blockIdx.x]);
}

// ---------------- Pass 2: deterministic final reduction --------------------
__global__ __launch_bounds__(256)
void k_finalize(const float* __restrict__ partials, float* __restrict__ totals,
                int nblk) {
  float sg = 0.0f, sh = 0.0f;
  for (int i = threadIdx.x; i < nblk; i += blockDim.x) {
    sg += partials[2 * i];
    sh += partials[2 * i + 1];
  }
  block_reduce_pair(sg, sh, totals);
}

// ---------------- Pass 3: elementwise output -------------------------------
__global__ __launch_bounds__(256)
void k_output(const float* __restrict__ g, const float* __restrict__ u,
              const float* __restrict__ cvec, const float* __restrict__ bu,
              const float* __restrict__ totals, float* __restrict__ out,
              int n4, int n) {
  // uniform scalar setup (compiler keeps this in SGPRs; loads hit L2)
  const float sumsq_g = totals[0];
  const float sumsq_h = totals[1];
  const float gnorm   = sqrtf(sumsq_g);
  float rho = sqrtf(sumsq_h) / (gnorm + EPSF);
  rho = fminf(fmaxf(rho, 0.01f), 8.0f);
  const bool  alert  = rho > RHO_THR;
  const float boost  = 4.0f + bu[0] * (11.0f - 4.0f);
  const float eff    = alert ? boost : 1.0f;
  const float nscale = (alert ? T_ENT : 0.0f) * gnorm * INV_SQRTN;
  const float cadd   = cvec[0] * SQC;

  const int tid    = blockIdx.x * blockDim.x + threadIdx.x;
  const int stride = gridDim.x * blockDim.x;
  const float4* g4 = (const float4*)g;
  const float4* u4 = (const float4*)u;
  float4*      o4  = (float4*)out;

  for (int i = tid; i < n4; i += stride) {
    const float4 gv = g4[i];
    const float4 uv = u4[i];
    float4 o;
    {
      const float nz = sqrtf(-2.0f * __logf(uv.x + 1e-10f)) * __cosf(TWO_PI * uv.x);
      o.x = gv.x * eff + nscale * (nz * SQ1MC + cadd);
    }
    {
      const float nz = sqrtf(-2.0f * __logf(uv.y + 1e-10f)) * __cosf(TWO_PI * uv.y);
      o.y = gv.y * eff + nscale * (nz * SQ1MC + cadd);
    }
    {
      const float nz = sqrtf(-2.0f * __logf(uv.z + 1e-10f)) * __cosf(TWO_PI * uv.z);
      o.z = gv.z * eff + nscale * (nz * SQ1MC + cadd);
    }
    {
      const float nz = sqrtf(-2.0f * __logf(uv.w + 1e-10f)) * __cosf(TWO_PI * uv.w);
      o.w = gv.w * eff + nscale * (nz * SQ1MC + cadd);
    }
    o4[i] = o;
  }
  for (int i = 4 * n4 + tid; i < n; i += stride) {
    const float uu = u[i];
    const float nz = sqrtf(-2.0f * __logf(uu + 1e-10f)) * __cosf(TWO_PI * uu);
    out[i] = g[i] * eff + nscale * (nz * SQ1MC + cadd);
  }
}

// ---------------------------------------------------------------------------
extern "C" void kernel_launch(void* const* d_in, const int* in_sizes, int n_in,
                              void* d_out, int out_size, void* d_ws, size_t ws_size,
                              hipStream_t stream) {
  const float* g_curr     = (const float*)d_in[0];
  const float* g_prev     = (const float*)d_in[1];
  const float* theta_curr = (const float*)d_in[2];
  const float* theta_prev = (const float*)d_in[3];
  const float* u          = (const float*)d_in[4];
  const float* c          = (const float*)d_in[5];
  const float* boost_u    = (const float*)d_in[6];
  float* out = (float*)d_out;

  const int n  = in_sizes[0];
  const int n4 = n / 4;

  float* wsf      = (float*)d_ws;
  float* partials = wsf;               // 2 * NBLK_RED floats (fully overwritten)
  float* totals   = wsf + TOTALS_OFF;  // 2 floats

  // Pass 1: partial sums of squares (g and hvp), one pair per block.
  k_reduce<<<NBLK_RED, 256, 0, stream>>>(g_curr, g_prev, theta_curr, theta_prev,
                                         partials, n4, n);
  // Pass 2: deterministic final reduce -> totals[0..1].
  k_finalize<<<1, 256, 0, stream>>>(partials, totals, NBLK_RED);
  // Pass 3: elementwise output.
  int oblocks = (n4 + 255) / 256;
  if (oblocks > 8192) oblocks = 8192;
  if (oblocks < 1)    oblocks = 1;
  k_output<<<oblocks, 256, 0, stream>>>(g_curr, u, c, boost_u, totals, out, n4, n);
}